// GAT_50757923504815
// MI455X (gfx1250) — compile-verified
//
#include <hip/hip_runtime.h>

#define NN 100000          // nodes
#define NE 1600000         // edges without self-loops
#define NET (NE + NN)      // edges including self-loops
#define NG 64              // graphs
#define FOUT3 32
#define NEG_SLOPE 0.2f

typedef __attribute__((ext_vector_type(2))) float v2f;
typedef __attribute__((ext_vector_type(8))) float v8f;

// ---------------------------------------------------------------- utilities

__device__ __forceinline__ void atomicMaxF(float* addr, float val) {
    // Monotonic float ordering via sign-split integer atomics.
    if (val >= 0.0f) atomicMax((int*)addr, __float_as_int(val));
    else             atomicMin((unsigned int*)addr, __float_as_uint(val));
}

__device__ __forceinline__ void edge_sd(const int* __restrict__ ei, int ee, int& s, int& d) {
    if (ee < NE) { s = ei[ee]; d = ei[NE + ee]; }
    else         { s = ee - NE; d = s; }        // synthesized self-loop
}

__global__ void fill_kernel(float* __restrict__ p, float v, int n) {
    int i = blockIdx.x * blockDim.x + threadIdx.x;
    if (i < n) p[i] = v;
}

// mm[i] = -FLT_MAX, dn[i] = 0 in one pass (buffers are adjacent: dn = mm + NN)
__global__ void init_softmax_kernel(float* __restrict__ mm) {
    int i = blockIdx.x * blockDim.x + threadIdx.x;
    if (i >= NN) return;
    mm[i]      = -3.402823466e38f;
    mm[NN + i] = 0.0f;
}

// ---------------------------------------------------------------- WMMA GEMM
// H[M,Fout] = X[M,K] @ W[K,Fout]; one 16x16 output tile per wave32 using
// V_WMMA_F32_16X16X4_F32 (full fp32 precision; GEMM is <<1% of runtime).
// A layout (ISA 7.12.2, 32-bit A 16x4): lanes 0-15 M=0..15 hold K=0,1;
// lanes 16-31 hold K=2,3. B mirrors with N across lanes. D VGPR r -> row
// r (lanes 0-15) / row r+8 (lanes 16-31), col = lane%16.
__global__ void __launch_bounds__(256)
gemm_wmma_kernel(const float* __restrict__ X, const float* __restrict__ W,
                 float* __restrict__ H, int K, int Fout, int tilesN, int totalTiles) {
    int wave = threadIdx.x >> 5;
    int lane = threadIdx.x & 31;
    int tile = blockIdx.x * (blockDim.x >> 5) + wave;
    if (tile >= totalTiles) return;              // wave-uniform: EXEC stays all-1s
    int m0  = (tile / tilesN) << 4;
    int n0  = (tile % tilesN) << 4;
    int hi  = lane >> 4;                         // 0: lanes 0-15, 1: lanes 16-31
    int l16 = lane & 15;
    const float* __restrict__ xrow = X + (size_t)(m0 + l16) * K;
    int col = n0 + l16;
    v8f c = {};
    #pragma unroll 4
    for (int k = 0; k < K; k += 4) {
        int kk = k + hi * 2;
        v2f a, b;
        a.x = xrow[kk];
        a.y = xrow[kk + 1];
        b.x = W[(size_t)kk * Fout + col];
        b.y = W[(size_t)(kk + 1) * Fout + col];
        c = __builtin_amdgcn_wmma_f32_16x16x4_f32(false, a, false, b,
                                                  (short)0, c, false, false);
    }
    int rowBase = m0 + hi * 8;
    #pragma unroll
    for (int r = 0; r < 8; ++r)
        H[(size_t)(rowBase + r) * Fout + col] = c[r];
}

// ---------------------------------------------------------------- per-node dots
__global__ void dots_kernel(const float* __restrict__ H, const float* __restrict__ av_s,
                            const float* __restrict__ av_d, float* __restrict__ hs,
                            float* __restrict__ hd, int n, int F) {
    int i = blockIdx.x * blockDim.x + threadIdx.x;
    if (i >= n) return;
    const float4* __restrict__ row = (const float4*)(H + (size_t)i * F);
    const float4* __restrict__ as4 = (const float4*)av_s;
    const float4* __restrict__ ad4 = (const float4*)av_d;
    float s = 0.f, d = 0.f;
    int c4 = F >> 2;
    for (int f = 0; f < c4; ++f) {
        float4 h = row[f], a = as4[f], b = ad4[f];
        s += h.x * a.x + h.y * a.y + h.z * a.z + h.w * a.w;
        d += h.x * b.x + h.y * b.y + h.z * b.z + h.w * b.w;
    }
    hs[i] = s; hd[i] = d;
}

// ---------------------------------------------------------------- edge phase
__global__ void edge_logits_kernel(const int* __restrict__ ei, const float* __restrict__ hs,
                                   const float* __restrict__ hd, float* __restrict__ ebuf,
                                   float* __restrict__ mmax) {
    int ee = blockIdx.x * blockDim.x + threadIdx.x;
    if (ee >= NET) return;
    int s, d; edge_sd(ei, ee, s, d);
    float e = hs[s] + hd[d];
    e = (e >= 0.f) ? e : NEG_SLOPE * e;          // leaky_relu
    ebuf[ee] = e;
    atomicMaxF(&mmax[d], e);
}

__global__ void edge_exp_kernel(const int* __restrict__ ei, float* __restrict__ ebuf,
                                const float* __restrict__ mmax, float* __restrict__ denom) {
    int ee = blockIdx.x * blockDim.x + threadIdx.x;
    if (ee >= NET) return;
    int s, d; edge_sd(ei, ee, s, d);
    float ex = __expf(ebuf[ee] - mmax[d]);
    ebuf[ee] = ex;
    atomicAdd(&denom[d], ex);
}

// One thread per (edge, 4-feature chunk): float4 gather from h[src], scaled by
// the UNnormalized exp weight; normalization by denom[dst] is folded into the
// node-wise epilogue (sum(ex*h)/denom == sum((ex/denom)*h)). All gather/atomic
// traffic resolves in the 192MB L2.
__global__ void edge_scatter_kernel(const int* __restrict__ ei, const float* __restrict__ ebuf,
                                    const float* __restrict__ H, float* __restrict__ agg,
                                    int F, int total) {
    int tid = blockIdx.x * blockDim.x + threadIdx.x;
    if (tid >= total) return;
    int chunks = F >> 2;
    int ee = tid / chunks;
    int ch = tid - ee * chunks;
    int s, d; edge_sd(ei, ee, s, d);
    float w = ebuf[ee];
    const float4 h = ((const float4*)(H + (size_t)s * F))[ch];
    float* o = agg + (size_t)d * F + (ch << 2);
    atomicAdd(o + 0, w * h.x);
    atomicAdd(o + 1, w * h.y);
    atomicAdd(o + 2, w * h.z);
    atomicAdd(o + 3, w * h.w);
}

// xout = relu(agg / (denom + 1e-16) + b)   [normalization folded here]
__global__ void norm_bias_relu_kernel(float* __restrict__ agg, const float* __restrict__ denom,
                                      const float* __restrict__ b, int n, int F) {
    int tid = blockIdx.x * blockDim.x + threadIdx.x;
    if (tid >= n * F) return;
    int i = tid / F;
    int f = tid - i * F;
    float v = agg[tid] / (denom[i] + 1e-16f) + b[f];
    agg[tid] = (v > 0.f) ? v : 0.f;
}

// ---------------------------------------------------------------- pooling
__global__ void pool_kernel(const float* __restrict__ x3, const int* __restrict__ batch,
                            float* __restrict__ pooled, float* __restrict__ cnt) {
    int tid = blockIdx.x * blockDim.x + threadIdx.x;
    int i = tid / FOUT3, f = tid - (tid / FOUT3) * FOUT3;
    if (i >= NN) return;
    int g = batch[i];
    atomicAdd(&pooled[g * FOUT3 + f], x3[(size_t)i * FOUT3 + f]);
    if (f == 0) atomicAdd(&cnt[g], 1.0f);
}

__global__ void pool_finalize_kernel(const float* __restrict__ pooled,
                                     const float* __restrict__ cnt, float* __restrict__ out) {
    int tid = blockIdx.x * blockDim.x + threadIdx.x;
    if (tid >= NG * FOUT3) return;
    out[tid] = pooled[tid] / fmaxf(cnt[tid / FOUT3], 1.0f);
}

// ---------------------------------------------------------------- driver

static void run_layer(const float* xin, int K, int F,
                      const float* W, const float* as_, const float* ad_, const float* b,
                      float* H, float* agg, const int* ei,
                      float* hs, float* hd, float* mm, float* dn, float* eb,
                      hipStream_t stream) {
    const int tilesN = F >> 4;
    const int totalTiles = (NN >> 4) * tilesN;
    gemm_wmma_kernel<<<(totalTiles + 7) / 8, 256, 0, stream>>>(xin, W, H, K, F, tilesN, totalTiles);
    dots_kernel<<<(NN + 255) / 256, 256, 0, stream>>>(H, as_, ad_, hs, hd, NN, F);
    fill_kernel<<<(NN * F + 255) / 256, 256, 0, stream>>>(agg, 0.0f, NN * F);
    init_softmax_kernel<<<(NN + 255) / 256, 256, 0, stream>>>(mm);   // mm=-FLT_MAX, dn=0
    edge_logits_kernel<<<(NET + 255) / 256, 256, 0, stream>>>(ei, hs, hd, eb, mm);
    edge_exp_kernel<<<(NET + 255) / 256, 256, 0, stream>>>(ei, eb, mm, dn);
    const int total = NET * (F >> 2);
    edge_scatter_kernel<<<(total + 255) / 256, 256, 0, stream>>>(ei, eb, H, agg, F, total);
    norm_bias_relu_kernel<<<(NN * F + 255) / 256, 256, 0, stream>>>(agg, dn, b, NN, F);
}

extern "C" void kernel_launch(void* const* d_in, const int* in_sizes, int n_in,
                              void* d_out, int out_size, void* d_ws, size_t ws_size,
                              hipStream_t stream) {
    (void)in_sizes; (void)n_in; (void)out_size; (void)ws_size;
    const float* x    = (const float*)d_in[0];
    const int*   ei   = (const int*)  d_in[1];
    const int*   batch= (const int*)  d_in[2];
    const float* W1   = (const float*)d_in[3];
    const float* as1  = (const float*)d_in[4];
    const float* ad1  = (const float*)d_in[5];
    const float* b1   = (const float*)d_in[6];
    const float* W2   = (const float*)d_in[7];
    const float* as2  = (const float*)d_in[8];
    const float* ad2  = (const float*)d_in[9];
    const float* b2   = (const float*)d_in[10];
    const float* W3   = (const float*)d_in[11];
    const float* as3  = (const float*)d_in[12];
    const float* ad3  = (const float*)d_in[13];
    const float* b3   = (const float*)d_in[14];
    float* out = (float*)d_out;

    // Workspace carve (floats): 3x [N,64] + 4x [N] + [E+N] + pooled/cnt  (~86 MB)
    float* ws   = (float*)d_ws;
    size_t NF   = (size_t)NN * 64;
    float* bufH = ws;                 // projected features h
    float* bufA = bufH + NF;          // layer output ping
    float* bufB = bufA + NF;          // layer output pong
    float* hs   = bufB + NF;
    float* hd   = hs + NN;
    float* mm   = hd + NN;            // segment max   (dn must follow mm!)
    float* dn   = mm + NN;            // segment sum of exp
    float* eb   = dn + NN;            // per-edge logits -> exp(e - m)
    float* pooled = eb + NET;         // [G,32] then [G] counts (contiguous)
    float* cnt    = pooled + NG * FOUT3;

    run_layer(x,    128, 64, W1, as1, ad1, b1, bufH, bufA, ei, hs, hd, mm, dn, eb, stream);
    run_layer(bufA,  64, 64, W2, as2, ad2, b2, bufH, bufB, ei, hs, hd, mm, dn, eb, stream);
    run_layer(bufB,  64, 32, W3, as3, ad3, b3, bufH, bufA, ei, hs, hd, mm, dn, eb, stream);

    fill_kernel<<<(NG * FOUT3 + NG + 255) / 256, 256, 0, stream>>>(pooled, 0.0f, NG * FOUT3 + NG);
    pool_kernel<<<(NN * FOUT3 + 255) / 256, 256, 0, stream>>>(bufA, batch, pooled, cnt);
    pool_finalize_kernel<<<(NG * FOUT3 + 255) / 256, 256, 0, stream>>>(pooled, cnt, out);
}